// AttnOnAttn_72885595013141
// MI455X (gfx1250) — compile-verified
//
#include <hip/hip_runtime.h>
#include <math.h>

typedef __attribute__((ext_vector_type(2))) float v2f;
typedef __attribute__((ext_vector_type(8))) float v8f;

#define N_ 8
#define L_ 512
#define EMB_ 320
#define H_ 20
#define RANK_ 20
#define CLIP_ 32

// ---- workspace layout (float offsets) ----
#define WS_X1   0
#define WS_X2   (WS_X1 + N_*L_*RANK_)          // 81920
#define WS_M1   (WS_X2 + N_*L_*RANK_)          // 163840
#define WS_G    (WS_M1 + 400)
#define WS_S    (WS_G + 400)
#define WS_POS  (WS_S + 200)
#define WS_SELB (WS_POS + 65*20)
#define WS_FC1  (WS_SELB + 10)
#define WS_FC1B (WS_FC1 + 20000)
#define WS_FC2  (WS_FC1B + 100)
#define WS_FC2B (WS_FC2 + 5000)
#define WS_FC3  (WS_FC2B + 50)
#define WS_FC3B (WS_FC3 + 50)
#define WS_SC1  (WS_FC3B + 1)
#define WS_SC2  (WS_SC1 + 1)

// ============================================================
// Kernel 1: weight norms + folded/derived weight matrices.
//   M1[o][p]  = Wl_n[p][o]                     (input-proj, pre-transposed)
//   G[r][p]   = sum_o Wo_n[o][r] * Wl_n[p][o]  (bilinear folded through Wl)
//   S[o][h]   = Ws_n[h][o]
//   POS[d][p] = Wp_n[p][d] + pos_b[p] + lin_b[p]   (biases folded)
// ============================================================
__global__ void prep_weights(
    const float* v1, const float* g1, const float* v2, const float* g2,
    const float* vo, const float* go,
    const float* lin_v, const float* lin_g, const float* lin_b,
    const float* pos_v, const float* pos_g, const float* pos_b,
    const float* sel_v, const float* sel_g, const float* sel_b,
    const float* f1v, const float* f1g, const float* f1b,
    const float* f2v, const float* f2g, const float* f2b,
    const float* f3v, const float* f3g, const float* f3b,
    float* ws)
{
  __shared__ float red[256];
  __shared__ float scl[9];
  const int tid = threadIdx.x;

  const float* mats[9] = {v1, v2, vo, lin_v, pos_v, sel_v, f1v, f2v, f3v};
  const int    cnts[9] = {RANK_*EMB_, RANK_*EMB_, 400, 400, 20*65, 200, 20000, 5000, 50};
  const float* gs[9]   = {g1, g2, go, lin_g, pos_g, sel_g, f1g, f2g, f3g};

  for (int w = 0; w < 9; ++w) {
    float s = 0.f;
    for (int idx = tid; idx < cnts[w]; idx += 256) { float xv = mats[w][idx]; s += xv*xv; }
    red[tid] = s; __syncthreads();
    for (int off = 128; off > 0; off >>= 1) {
      if (tid < off) red[tid] += red[tid+off];
      __syncthreads();
    }
    if (tid == 0) scl[w] = gs[w][0] / sqrtf(red[0]);
    __syncthreads();
  }

  const float so = scl[2], sl = scl[3], sp = scl[4], ss = scl[5];
  for (int idx = tid; idx < 400; idx += 256) {
    int a = idx / 20, p = idx % 20;              // a = o (for M1), a = r (for G)
    ws[WS_M1 + idx] = lin_v[p*20 + a] * sl;
    float g = 0.f;
    for (int o = 0; o < 20; ++o) g += vo[o*20 + a] * lin_v[p*20 + o];
    ws[WS_G + idx] = g * so * sl;
  }
  for (int idx = tid; idx < 200; idx += 256) {
    int o = idx / 10, h = idx % 10;
    ws[WS_S + idx] = sel_v[h*20 + o] * ss;
  }
  for (int idx = tid; idx < 65*20; idx += 256) {
    int d = idx / 20, p = idx % 20;
    ws[WS_POS + idx] = pos_v[p*65 + d] * sp + pos_b[p] + lin_b[p];
  }
  if (tid < 10) ws[WS_SELB + tid] = sel_b[tid];

  const float sf1 = scl[6], sf2 = scl[7], sf3 = scl[8];
  for (int idx = tid; idx < 20000; idx += 256) ws[WS_FC1 + idx] = f1v[idx] * sf1;
  for (int idx = tid; idx < 100;   idx += 256) ws[WS_FC1B + idx] = f1b[idx];
  for (int idx = tid; idx < 5000;  idx += 256) ws[WS_FC2 + idx] = f2v[idx] * sf2;
  for (int idx = tid; idx < 50;    idx += 256) {
    ws[WS_FC2B + idx] = f2b[idx];
    ws[WS_FC3 + idx]  = f3v[idx] * sf3;
  }
  if (tid == 0) { ws[WS_FC3B] = f3b[0]; ws[WS_SC1] = scl[0]; ws[WS_SC2] = scl[1]; }
}

// ============================================================
// Kernel 2: x1/x2 = emb @ Wn{1,2}.T  (per sequence position)
// ============================================================
__global__ void prep_x12(const float* __restrict__ emb,
                         const float* __restrict__ v1,
                         const float* __restrict__ v2,
                         float* __restrict__ ws)
{
  const int b = blockIdx.x;                 // n*L + i
  const int tid = threadIdx.x;
  const float s1 = ws[WS_SC1], s2 = ws[WS_SC2];
  const float* e = emb + (size_t)b * EMB_;
  if (tid < RANK_) {
    const float* w = v1 + tid * EMB_;
    float acc = 0.f;
    for (int k = 0; k < EMB_; ++k) acc += e[k] * w[k];
    ws[WS_X1 + (size_t)b * RANK_ + tid] = acc * s1;
  } else if (tid >= 32 && tid < 32 + RANK_) {
    const int r = tid - 32;
    const float* w = v2 + r * EMB_;
    float acc = 0.f;
    for (int k = 0; k < EMB_; ++k) acc += e[k] * w[k];
    ws[WS_X2 + (size_t)b * RANK_ + r] = acc * s2;
  }
}

// ============================================================
// Kernel 3: fused main pass, one workgroup per (n,i) row.
//   T[j,:] = x[n,i,j,:]@M1 + x2[n,j,:]@(diag(x1_i)*G) + POS[idx(i,j)]
//   s = T@S + selb -> softmax over j -> sv[h,o] = sum_j v[j,h]T[j,o]
//   out = fc3(relu(fc2(relu(fc1(sv)))))
// ============================================================
#define TSTR 21

__global__ __launch_bounds__(256) void attn_main(const float* __restrict__ x,
                                                 const float* __restrict__ ws,
                                                 float* __restrict__ out)
{
  extern __shared__ float sm[];
  float* T     = sm;                 // 512*21 = 10752
  float* sArr  = sm + 512*TSTR;      // 512*10 =  5120
  float* lM1   = sArr + 512*10;      // 400
  float* lM2   = lM1 + 400;          // 400
  float* lS    = lM2 + 400;          // 200
  float* lPos  = lS + 200;           // 1300
  float* lSelb = lPos + 1300;        // 10
  float* maxh  = lSelb + 10;         // 10
  float* denom = maxh + 10;          // 10
  float* pred  = denom + 10;         // 160
  float* hvec  = pred + 160;         // 200
  float* h1    = hvec + 200;         // 100
  float* h2    = h1 + 100;           // 50

  const int tid = threadIdx.x;
  const int b   = blockIdx.x;
  const int n   = b >> 9;
  const int i   = b & (L_ - 1);

  // ---- phase 0: stage weights; build per-row M2 = diag(x1_i) * G ----
  const float* x1row = ws + WS_X1 + (size_t)b * RANK_;
  for (int idx = tid; idx < 400; idx += 256) {
    lM1[idx] = ws[WS_M1 + idx];
    lM2[idx] = x1row[idx / 20] * ws[WS_G + idx];
  }
  for (int idx = tid; idx < 200;  idx += 256) lS[idx]   = ws[WS_S + idx];
  for (int idx = tid; idx < 1300; idx += 256) lPos[idx] = ws[WS_POS + idx];
  if (tid < 10) lSelb[tid] = ws[WS_SELB + tid];
  __syncthreads();

  // ---- phase 1: T via V_WMMA_F32_16X16X4_F32 ----
  const int lane  = tid & 31;
  const int wv    = tid >> 5;          // wave id, 8 waves
  const int m     = lane & 15;
  const int hi    = lane >> 4;         // 0 or 1
  const int k0sel = hi << 1;           // A/B K-offset for this half-wave
  const float* xrow   = x + (size_t)b * L_ * H_;
  const float* x2base = ws + WS_X2 + (size_t)n * L_ * RANK_;

  // B fragments are tile-invariant: load once into registers (frees the hot
  // loop from ds_load + s_wait_dscnt serialization against the WMMA chain).
  v2f b1f[2][5], b2f[2][5];
#pragma unroll
  for (int nt = 0; nt < 2; ++nt) {
    const int p = (lane & 15) + nt * 16;
    const bool pv = (p < 20);
#pragma unroll
    for (int kc = 0; kc < 5; ++kc) {
      const int k0 = kc * 4 + k0sel;
      b1f[nt][kc].x = pv ? lM1[k0*20 + p]     : 0.f;
      b1f[nt][kc].y = pv ? lM1[(k0+1)*20 + p] : 0.f;
      b2f[nt][kc].x = pv ? lM2[k0*20 + p]     : 0.f;
      b2f[nt][kc].y = pv ? lM2[(k0+1)*20 + p] : 0.f;
    }
  }

  for (int t = 0; t < 4; ++t) {        // 4 M-tiles per wave (32 total)
    const int jb = (wv * 4 + t) * 16;
    const int j  = jb + m;
    const float* xp  = xrow   + j * 20;
    const float* x2p = x2base + j * 20;
    v2f ax[5], a2[5];
#pragma unroll
    for (int kc = 0; kc < 5; ++kc) {   // K = 20 in 5 chunks of 4
      const int k0 = kc * 4 + k0sel;
      ax[kc].x = xp[k0];  ax[kc].y = xp[k0+1];
      a2[kc].x = x2p[k0]; a2[kc].y = x2p[k0+1];
    }
#pragma unroll
    for (int nt = 0; nt < 2; ++nt) {   // 20 output cols padded to 2 N-tiles
      const int p = (lane & 15) + nt * 16;
      // Two independent accumulation chains -> half the WMMA RAW depth.
      v8f accA = {0.f,0.f,0.f,0.f,0.f,0.f,0.f,0.f};
      v8f accB = {0.f,0.f,0.f,0.f,0.f,0.f,0.f,0.f};
#pragma unroll
      for (int kc = 0; kc < 5; ++kc) {
        accA = __builtin_amdgcn_wmma_f32_16x16x4_f32(false, ax[kc], false, b1f[nt][kc],
                                                     (short)0, accA, false, false);
        accB = __builtin_amdgcn_wmma_f32_16x16x4_f32(false, a2[kc], false, b2f[nt][kc],
                                                     (short)0, accB, false, false);
      }
      if (p < 20) {
#pragma unroll
        for (int v = 0; v < 8; ++v) {
          const int jr = jb + v + hi * 8;           // C/D layout row
          int d = jr - i;
          d = (d < -CLIP_) ? -CLIP_ : (d > CLIP_ ? CLIP_ : d);
          T[jr * TSTR + p] = accA[v] + accB[v] + lPos[(d + CLIP_) * 20 + p];
        }
      }
    }
  }
  __syncthreads();

  // ---- phase 2: selector logits + softmax over j ----
  for (int j = tid; j < L_; j += 256) {
    float sv[10];
#pragma unroll
    for (int h = 0; h < 10; ++h) sv[h] = lSelb[h];
#pragma unroll
    for (int o = 0; o < 20; ++o) {
      const float tv = T[j * TSTR + o];
#pragma unroll
      for (int h = 0; h < 10; ++h) sv[h] += tv * lS[o*10 + h];
    }
#pragma unroll
    for (int h = 0; h < 10; ++h) sArr[j*10 + h] = sv[h];
  }
  __syncthreads();
  if (tid < 160) {
    const int h = tid % 10, c = tid / 10;
    float mx = -3.4e38f;
    for (int j = c*32; j < c*32 + 32; ++j) mx = fmaxf(mx, sArr[j*10 + h]);
    pred[c*10 + h] = mx;
  }
  __syncthreads();
  if (tid < 10) {
    float mx = -3.4e38f;
    for (int c = 0; c < 16; ++c) mx = fmaxf(mx, pred[c*10 + tid]);
    maxh[tid] = mx;
  }
  __syncthreads();
  if (tid < 160) {
    const int h = tid % 10, c = tid / 10;
    const float mh = maxh[h];
    float s = 0.f;
    for (int j = c*32; j < c*32 + 32; ++j) {
      const float e = __expf(sArr[j*10 + h] - mh);
      sArr[j*10 + h] = e;
      s += e;
    }
    pred[c*10 + h] = s;
  }
  __syncthreads();
  if (tid < 10) {
    float s = 0.f;
    for (int c = 0; c < 16; ++c) s += pred[c*10 + tid];
    denom[tid] = s;
  }
  __syncthreads();

  // ---- phase 3: sv[h,o] = (1/denom[h]) * sum_j e[j,h]*T[j,o] ----
  if (tid < 200) {
    const int h = tid / 20, o = tid % 20;
    float acc = 0.f;
    for (int j = 0; j < L_; ++j) acc += sArr[j*10 + h] * T[j*TSTR + o];
    hvec[tid] = acc / denom[h];
  }
  __syncthreads();

  // ---- phase 4: FC head (weights L2-resident) ----
  if (tid < 100) {
    const float* w = ws + WS_FC1 + tid * 200;
    float acc = ws[WS_FC1B + tid];
    for (int c = 0; c < 200; ++c) acc += hvec[c] * w[c];
    h1[tid] = fmaxf(acc, 0.f);
  }
  __syncthreads();
  if (tid < 50) {
    const float* w = ws + WS_FC2 + tid * 100;
    float acc = ws[WS_FC2B + tid];
    for (int c = 0; c < 100; ++c) acc += h1[c] * w[c];
    h2[tid] = fmaxf(acc, 0.f);
  }
  __syncthreads();
  if (tid == 0) {
    float acc = ws[WS_FC3B];
    for (int c = 0; c < 50; ++c) acc += h2[c] * ws[WS_FC3 + c];
    out[b] = acc;
  }
}

// ============================================================
extern "C" void kernel_launch(void* const* d_in, const int* in_sizes, int n_in,
                              void* d_out, int out_size, void* d_ws, size_t ws_size,
                              hipStream_t stream)
{
  const float* x    = (const float*)d_in[0];
  const float* emb  = (const float*)d_in[1];
  const float* v1   = (const float*)d_in[2];
  const float* g1   = (const float*)d_in[3];
  const float* v2   = (const float*)d_in[4];
  const float* g2   = (const float*)d_in[5];
  const float* vo   = (const float*)d_in[6];
  const float* go   = (const float*)d_in[7];
  const float* linv = (const float*)d_in[8];
  const float* ling = (const float*)d_in[9];
  const float* linb = (const float*)d_in[10];
  const float* posv = (const float*)d_in[11];
  const float* posg = (const float*)d_in[12];
  const float* posb = (const float*)d_in[13];
  const float* selv = (const float*)d_in[14];
  const float* selg = (const float*)d_in[15];
  const float* selb = (const float*)d_in[16];
  const float* f1v  = (const float*)d_in[17];
  const float* f1g  = (const float*)d_in[18];
  const float* f1b  = (const float*)d_in[19];
  const float* f2v  = (const float*)d_in[20];
  const float* f2g  = (const float*)d_in[21];
  const float* f2b  = (const float*)d_in[22];
  const float* f3v  = (const float*)d_in[23];
  const float* f3g  = (const float*)d_in[24];
  const float* f3b  = (const float*)d_in[25];
  float* ws  = (float*)d_ws;
  float* out = (float*)d_out;

  hipLaunchKernelGGL(prep_weights, dim3(1), dim3(256), 0, stream,
      v1, g1, v2, g2, vo, go, linv, ling, linb, posv, posg, posb,
      selv, selg, selb, f1v, f1g, f1b, f2v, f2g, f2b, f3v, f3g, f3b, ws);

  hipLaunchKernelGGL(prep_x12, dim3(N_ * L_), dim3(64), 0, stream, emb, v1, v2, ws);

  const size_t shmem = (size_t)(512*TSTR + 512*10 + 400 + 400 + 200 + 1300 +
                                10 + 10 + 10 + 160 + 200 + 100 + 50) * sizeof(float);
  hipLaunchKernelGGL(attn_main, dim3(N_ * L_), dim3(256), shmem, stream, x, ws, out);
}